// HierarchicalSoftmax_94489281221
// MI455X (gfx1250) — compile-verified
//
#include <hip/hip_runtime.h>
#include <math.h>

#define NHID      1024
#define TPC       225
#define NCLASSES  224
#define BATCH     2048
#define NT_BOT    15          // ceil(225/16)
#define BLG_PITCH 240         // padded bottom-logit row stride

typedef __attribute__((ext_vector_type(2))) float v2f;
typedef __attribute__((ext_vector_type(8))) float v8f;

// ---------------------------------------------------------------------------
// Kernel 1: decompose labels, histogram classes, exclusive scan, scatter perm
// ---------------------------------------------------------------------------
__global__ __launch_bounds__(256) void bucket_kernel(
    const int* __restrict__ labels,
    int* __restrict__ cls, int* __restrict__ word,
    int* __restrict__ perm, int* __restrict__ off) {
  __shared__ int s_cnt[NCLASSES];
  __shared__ int s_fill[NCLASSES];
  __shared__ int s_off[NCLASSES + 1];
  int tid = threadIdx.x;
  for (int c = tid; c < NCLASSES; c += 256) { s_cnt[c] = 0; s_fill[c] = 0; }
  __syncthreads();
  for (int i = tid; i < BATCH; i += 256) {
    int lab = labels[i];
    int c = lab / TPC;
    cls[i]  = c;
    word[i] = lab - c * TPC;
    atomicAdd(&s_cnt[c], 1);
  }
  __syncthreads();
  if (tid == 0) {
    int run = 0;
    for (int c = 0; c < NCLASSES; ++c) { s_off[c] = run; run += s_cnt[c]; }
    s_off[NCLASSES] = run;           // == BATCH
  }
  __syncthreads();
  for (int i = tid; i < BATCH; i += 256) {
    int c = cls[i];
    int r = atomicAdd(&s_fill[c], 1);
    perm[s_off[c] + r] = i;
  }
  for (int c = tid; c <= NCLASSES; c += 256) off[c] = s_off[c];
}

// ---------------------------------------------------------------------------
// Kernel 2: top-level GEMM  TL = X @ W_top + b_top   via V_WMMA_F32_16X16X4_F32
//   one wave per 16x16 tile; grid (128, 14); both dims divide exactly.
// ---------------------------------------------------------------------------
__global__ __launch_bounds__(32) void top_gemm_kernel(
    const float* __restrict__ X,      // [BATCH, NHID]
    const float* __restrict__ Wt,     // [NHID, NCLASSES]
    const float* __restrict__ bt,     // [NCLASSES]
    float* __restrict__ TL) {         // [BATCH, NCLASSES]
  const int mt = blockIdx.x, nt = blockIdx.y;
  const int lane  = threadIdx.x;
  const int half  = lane >> 4;        // 0: K pair {0,1}, 1: K pair {2,3}
  const int lan16 = lane & 15;
  const int m = mt * 16 + lan16;      // A row held by this lane
  const int n = nt * 16 + lan16;      // B/D column held by this lane
  const float* __restrict__ xrow = X + (size_t)m * NHID;

  v8f acc = {};
  for (int k = 0; k < NHID; k += 4) {
    const int ka = k + 2 * half;
    v2f a, b;
    a.x = xrow[ka];
    a.y = xrow[ka + 1];
    b.x = Wt[(size_t)ka * NCLASSES + n];
    b.y = Wt[(size_t)(ka + 1) * NCLASSES + n];
    acc = __builtin_amdgcn_wmma_f32_16x16x4_f32(
        false, a, false, b, (short)0, acc, false, false);
  }
  const float bias = bt[n];
#pragma unroll
  for (int q = 0; q < 8; ++q) {
    int row = mt * 16 + q + half * 8;  // C/D layout: VGPR q, lanes>=16 -> M+8
    TL[(size_t)row * NCLASSES + n] = acc[q] + bias;
  }
}

// ---------------------------------------------------------------------------
// Kernel 3: per-row softmax over 224 classes -> p_cls[e]  (one wave per row)
// ---------------------------------------------------------------------------
__global__ __launch_bounds__(32) void top_softmax_kernel(
    const float* __restrict__ TL, const int* __restrict__ cls,
    float* __restrict__ pcls) {
  const int e = blockIdx.x;
  const int lane = threadIdx.x;
  const float* __restrict__ row = TL + (size_t)e * NCLASSES;
  float mx = -3.0e38f;
  for (int nidx = lane; nidx < NCLASSES; nidx += 32) mx = fmaxf(mx, row[nidx]);
#pragma unroll
  for (int o = 16; o > 0; o >>= 1) mx = fmaxf(mx, __shfl_xor(mx, o, 32));
  float s = 0.f;
  for (int nidx = lane; nidx < NCLASSES; nidx += 32) s += __expf(row[nidx] - mx);
#pragma unroll
  for (int o = 16; o > 0; o >>= 1) s += __shfl_xor(s, o, 32);
  if (lane == 0) pcls[e] = __expf(row[cls[e]] - mx) / s;
}

// ---------------------------------------------------------------------------
// Kernel 4: bottom-level class-grouped GEMM via WMMA.
//   grid (NCLASSES, NT_BOT, 2); wave computes 16 examples x 16 words.
//   All indices clamped so every load is in-bounds and EXEC stays all-ones
//   around the WMMA; invalid rows/cols only pollute unstored D entries.
// ---------------------------------------------------------------------------
__global__ __launch_bounds__(32) void bottom_gemm_kernel(
    const float* __restrict__ X,      // [BATCH, NHID]
    const float* __restrict__ Wb,     // [NCLASSES, NHID, TPC]
    const float* __restrict__ bb,     // [NCLASSES, TPC]
    const int* __restrict__ perm, const int* __restrict__ off,
    float* __restrict__ BLG) {        // [BATCH, BLG_PITCH]
  const int c  = blockIdx.x;
  const int nt = blockIdx.y;
  const int lane  = threadIdx.x;
  const int half  = lane >> 4;
  const int lan16 = lane & 15;
  const int start = off[c];
  const int nc    = off[c + 1] - start;    // wave-uniform
  const int n     = nt * 16 + lan16;
  const int nn    = (n < TPC) ? n : (TPC - 1);   // clamped, in-bounds
  const float* __restrict__ W = Wb + (size_t)c * NHID * TPC;

  for (int mt = blockIdx.z; mt * 16 < nc; mt += gridDim.z) {
    const int r  = mt * 16 + lan16;
    const int rr = (r < nc) ? r : (nc - 1);       // clamped, in-bounds
    const int e  = perm[start + rr];
    const float* __restrict__ xrow = X + (size_t)e * NHID;

    v8f acc = {};
    for (int k = 0; k < NHID; k += 4) {
      const int ka = k + 2 * half;
      v2f a, b;
      a.x = xrow[ka];
      a.y = xrow[ka + 1];
      b.x = W[(size_t)ka * TPC + nn];
      b.y = W[(size_t)(ka + 1) * TPC + nn];
      acc = __builtin_amdgcn_wmma_f32_16x16x4_f32(
          false, a, false, b, (short)0, acc, false, false);
    }
    const float bias = bb[(size_t)c * TPC + nn];
#pragma unroll
    for (int q = 0; q < 8; ++q) {
      int row = mt * 16 + q + half * 8;
      if (row < nc && n < TPC)
        BLG[(size_t)(start + row) * BLG_PITCH + n] = acc[q] + bias;
    }
  }
}

// ---------------------------------------------------------------------------
// Kernel 5: bottom softmax over 225 words + final product (one wave per row)
// ---------------------------------------------------------------------------
__global__ __launch_bounds__(32) void bottom_final_kernel(
    const float* __restrict__ BLG, const int* __restrict__ perm,
    const int* __restrict__ word, const float* __restrict__ pcls,
    float* __restrict__ out) {
  const int p = blockIdx.x;
  const int lane = threadIdx.x;
  const int e = perm[p];
  const float* __restrict__ row = BLG + (size_t)p * BLG_PITCH;
  float mx = -3.0e38f;
  for (int nidx = lane; nidx < TPC; nidx += 32) mx = fmaxf(mx, row[nidx]);
#pragma unroll
  for (int o = 16; o > 0; o >>= 1) mx = fmaxf(mx, __shfl_xor(mx, o, 32));
  float s = 0.f;
  for (int nidx = lane; nidx < TPC; nidx += 32) s += __expf(row[nidx] - mx);
#pragma unroll
  for (int o = 16; o > 0; o >>= 1) s += __shfl_xor(s, o, 32);
  if (lane == 0) {
    float pw = __expf(row[word[e]] - mx) / s;
    out[e] = pcls[e] * pw;
  }
}

// ---------------------------------------------------------------------------
extern "C" void kernel_launch(void* const* d_in, const int* in_sizes, int n_in,
                              void* d_out, int out_size, void* d_ws, size_t ws_size,
                              hipStream_t stream) {
  const float* X      = (const float*)d_in[0];   // [2048,1024]
  const int*   labels = (const int*)  d_in[1];   // [2048]
  const float* Wt     = (const float*)d_in[2];   // [1024,224]
  const float* bt     = (const float*)d_in[3];   // [224]
  const float* Wb     = (const float*)d_in[4];   // [224,1024,225]
  const float* bb     = (const float*)d_in[5];   // [224,225]
  float* out = (float*)d_out;

  char* base = (char*)d_ws;
  int*   cls  = (int*)(base + 0);                        //  8 KB
  int*   word = (int*)(base + 8 * 1024);                 //  8 KB
  int*   perm = (int*)(base + 16 * 1024);                //  8 KB
  int*   off  = (int*)(base + 24 * 1024);                //  1 KB (225 ints)
  float* pcls = (float*)(base + 28 * 1024);              //  8 KB
  float* TL   = (float*)(base + 64 * 1024);              // 2048*224*4 = 1.75 MB
  float* BLG  = (float*)(base + 64 * 1024 + (size_t)BATCH * NCLASSES * 4);
                                                         // 2048*240*4 = 1.875 MB

  bucket_kernel<<<1, 256, 0, stream>>>(labels, cls, word, perm, off);
  top_gemm_kernel<<<dim3(BATCH / 16, NCLASSES / 16), 32, 0, stream>>>(X, Wt, bt, TL);
  top_softmax_kernel<<<BATCH, 32, 0, stream>>>(TL, cls, pcls);
  bottom_gemm_kernel<<<dim3(NCLASSES, NT_BOT, 2), 32, 0, stream>>>(X, Wb, bb, perm, off, BLG);
  bottom_final_kernel<<<BATCH, 32, 0, stream>>>(BLG, perm, word, pcls, out);
}